// MDNNModel_5_4784593568255
// MI455X (gfx1250) — compile-verified
//
#include <hip/hip_runtime.h>
#include <math.h>

#define N_NODES   100000
#define N_EDGES   640000
#define IN_FEATS  256
#define HEADS     4
#define OUT_FEATS 32
#define HD        (HEADS * OUT_FEATS)   // 128
#define NEG_SLOPE 0.2f

typedef __attribute__((ext_vector_type(2))) float v2f;
typedef __attribute__((ext_vector_type(8))) float v8f;

// ---------------------------------------------------------------------------
// Init: rst (output accumulator) = 0, m = -inf, denom = 0
// ---------------------------------------------------------------------------
__global__ void gat_init(float* __restrict__ m, float* __restrict__ denom,
                         float* __restrict__ rst) {
    const long total = (long)N_NODES * HD;           // 12.8M
    for (long i = (long)blockIdx.x * blockDim.x + threadIdx.x; i < total;
         i += (long)gridDim.x * blockDim.x) {
        rst[i] = 0.0f;
        if (i < (long)N_NODES * HEADS) {
            m[i] = -INFINITY;
            denom[i] = 0.0f;
        }
    }
}

// ---------------------------------------------------------------------------
// GEMM: ft[N,128] = feat[N,256] @ W[256,128] using V_WMMA_F32_16X16X4_F32
// One block = 16 rows of feat. 8 waves; wave w owns 16-col tile w.
// A strip staged in LDS with padded stride (260) to avoid bank conflicts.
// ---------------------------------------------------------------------------
#define LDS_STRIDE 260

__global__ __launch_bounds__(256) void gat_gemm_wmma(
    const float* __restrict__ feat, const float* __restrict__ W,
    float* __restrict__ ft) {
    __shared__ float lA[16 * LDS_STRIDE];

    const int r0  = blockIdx.x * 16;
    const int tid = threadIdx.x;

    // Cooperative load of A strip: 16 rows x 256 cols, as float4 (1024 chunks)
    for (int i = tid; i < 16 * 64; i += 256) {
        const int row = i >> 6;
        const int c4  = i & 63;
        const float4 v =
            *(const float4*)(feat + (size_t)(r0 + row) * IN_FEATS + c4 * 4);
        *(float4*)(lA + row * LDS_STRIDE + c4 * 4) = v;
    }
    __syncthreads();

    const int wave = tid >> 5;        // 0..7 -> column tile
    const int lane = tid & 31;
    const int half = lane >> 4;       // 0 or 1
    const int n16  = lane & 15;
    const int col0 = wave * 16;

    const float* __restrict__ Arow = lA + n16 * LDS_STRIDE;

    v8f acc = {};
    #pragma unroll 8
    for (int k = 0; k < IN_FEATS; k += 4) {
        const int ka = k + 2 * half;  // lanes 0-15: K=k..k+1, lanes 16-31: K=k+2..k+3
        v2f a;
        a.x = Arow[ka];
        a.y = Arow[ka + 1];
        v2f b;
        b.x = W[(size_t)ka * HD + col0 + n16];
        b.y = W[(size_t)(ka + 1) * HD + col0 + n16];
        acc = __builtin_amdgcn_wmma_f32_16x16x4_f32(
            /*neg_a=*/false, a, /*neg_b=*/false, b,
            /*c_mod=*/(short)0, acc, /*reuse_a=*/false, /*reuse_b=*/false);
    }

    // C layout: VGPR v -> row (v + 8*half), col = n16
    #pragma unroll
    for (int v = 0; v < 8; ++v) {
        const int row = r0 + v + 8 * half;
        ft[(size_t)row * HD + col0 + n16] = acc[v];
    }
}

// ---------------------------------------------------------------------------
// Per-node attention logits: el[n,h] = dot(ft[n,h,:], attn_l[h,:]), same for er
// ---------------------------------------------------------------------------
__global__ void gat_attn_logits(const float* __restrict__ ft,
                                const float* __restrict__ attn_l,
                                const float* __restrict__ attn_r,
                                float* __restrict__ el,
                                float* __restrict__ er) {
    const int idx = blockIdx.x * blockDim.x + threadIdx.x;  // n*HEADS + h
    if (idx >= N_NODES * HEADS) return;
    const int n = idx / HEADS;
    const int h = idx % HEADS;
    const float4* f  = (const float4*)(ft + (size_t)n * HD + h * OUT_FEATS);
    const float4* al = (const float4*)(attn_l + h * OUT_FEATS);
    const float4* ar = (const float4*)(attn_r + h * OUT_FEATS);
    float sl = 0.0f, sr = 0.0f;
    #pragma unroll
    for (int i = 0; i < OUT_FEATS / 4; ++i) {
        const float4 v = f[i], a = al[i], b = ar[i];
        sl += v.x * a.x + v.y * a.y + v.z * a.z + v.w * a.w;
        sr += v.x * b.x + v.y * b.y + v.z * b.z + v.w * b.w;
    }
    el[idx] = sl;
    er[idx] = sr;
}

// float atomic max via monotone int/uint bit ordering
__device__ __forceinline__ void atomicMaxFloat(float* addr, float value) {
    if (value >= 0.0f)
        atomicMax((int*)addr, __float_as_int(value));
    else
        atomicMin((unsigned int*)addr, __float_as_uint(value));
}

// ---------------------------------------------------------------------------
// Edge pass 1: e = leakyrelu(el[src] + er[dst]); segment max into m[dst]
// ---------------------------------------------------------------------------
__global__ void gat_edge_logits(const float* __restrict__ el,
                                const float* __restrict__ er,
                                const int* __restrict__ src,
                                const int* __restrict__ dst,
                                float* __restrict__ elog,
                                float* __restrict__ m) {
    const int idx = blockIdx.x * blockDim.x + threadIdx.x;  // eid*HEADS + h
    if (idx >= N_EDGES * HEADS) return;
    const int eid = idx / HEADS;
    const int h   = idx % HEADS;
    const int s = src[eid];
    const int d = dst[eid];
    float v = el[s * HEADS + h] + er[d * HEADS + h];
    v = v > 0.0f ? v : NEG_SLOPE * v;
    elog[idx] = v;
    atomicMaxFloat(&m[d * HEADS + h], v);
}

// ---------------------------------------------------------------------------
// Edge pass 2: ex = exp(e - m[dst]); segment sum into denom[dst]
// ---------------------------------------------------------------------------
__global__ void gat_edge_exp(const int* __restrict__ dst,
                             const float* __restrict__ m,
                             float* __restrict__ elog,
                             float* __restrict__ denom) {
    const int idx = blockIdx.x * blockDim.x + threadIdx.x;
    if (idx >= N_EDGES * HEADS) return;
    const int eid = idx / HEADS;
    const int h   = idx % HEADS;
    const int d   = dst[eid];
    const float ex = __expf(elog[idx] - m[d * HEADS + h]);
    elog[idx] = ex;
    atomicAdd(&denom[d * HEADS + h], ex);
}

// ---------------------------------------------------------------------------
// Edge pass 3: a = ex/denom[dst]; write a; rst[dst,h,:] += a * ft[src,h,:]
// One 128-thread block per edge: thread t -> (h = t/32, d = t%32).
// ft and rst are both L2-resident (51 MB each, 192 MB L2).
// ---------------------------------------------------------------------------
__global__ __launch_bounds__(128) void gat_aggregate(
    const int* __restrict__ src, const int* __restrict__ dst,
    const float* __restrict__ ft, const float* __restrict__ elog,
    const float* __restrict__ denom, float* __restrict__ rst,
    float* __restrict__ a_out) {
    const int eid = blockIdx.x;
    const int t   = threadIdx.x;
    const int h   = t >> 5;
    const int dch = t & 31;
    const int s = src[eid];
    const int d = dst[eid];
    const float ex = elog[eid * HEADS + h];
    const float a  = ex / denom[d * HEADS + h];
    if (dch == 0) a_out[eid * HEADS + h] = a;
    const float val = a * ft[(size_t)s * HD + h * OUT_FEATS + dch];
    atomicAdd(&rst[(size_t)d * HD + h * OUT_FEATS + dch], val);
}

// ---------------------------------------------------------------------------
extern "C" void kernel_launch(void* const* d_in, const int* in_sizes, int n_in,
                              void* d_out, int out_size, void* d_ws,
                              size_t ws_size, hipStream_t stream) {
    const float* feat   = (const float*)d_in[0];
    const float* W      = (const float*)d_in[1];
    const float* attn_l = (const float*)d_in[2];
    const float* attn_r = (const float*)d_in[3];
    const int*   src    = (const int*)d_in[4];
    const int*   dst    = (const int*)d_in[5];

    float* rst   = (float*)d_out;                        // [N, H, D]
    float* a_out = (float*)d_out + (size_t)N_NODES * HD; // [E, H, 1]

    float* ws    = (float*)d_ws;
    float* ft    = ws;                                   // N*HD
    float* el    = ft + (size_t)N_NODES * HD;            // N*H
    float* er    = el + (size_t)N_NODES * HEADS;         // N*H
    float* m     = er + (size_t)N_NODES * HEADS;         // N*H
    float* denom = m + (size_t)N_NODES * HEADS;          // N*H
    float* elog  = denom + (size_t)N_NODES * HEADS;      // E*H

    gat_init<<<8192, 256, 0, stream>>>(m, denom, rst);

    gat_gemm_wmma<<<N_NODES / 16, 256, 0, stream>>>(feat, W, ft);

    gat_attn_logits<<<(N_NODES * HEADS + 255) / 256, 256, 0, stream>>>(
        ft, attn_l, attn_r, el, er);

    gat_edge_logits<<<(N_EDGES * HEADS + 255) / 256, 256, 0, stream>>>(
        el, er, src, dst, elog, m);

    gat_edge_exp<<<(N_EDGES * HEADS + 255) / 256, 256, 0, stream>>>(
        dst, m, elog, denom);

    gat_aggregate<<<N_EDGES, 128, 0, stream>>>(src, dst, ft, elog, denom, rst,
                                               a_out);
}